// GCNLinkPredictor_51934744543384
// MI455X (gfx1250) — compile-verified
//
#include <hip/hip_runtime.h>
#include <hip/hip_bf16.h>
#include <math.h>

typedef float v2f __attribute__((ext_vector_type(2)));
typedef float v8f __attribute__((ext_vector_type(8)));

#define F 128

// ---------------- degree kernels ----------------
__global__ void deg_init(float* deg, int N) {
    int i = blockIdx.x * blockDim.x + threadIdx.x;
    if (i < N) deg[i] = 1.0f;  // self-loop
}

__global__ void deg_count(const long long* __restrict__ dst, float* deg, int E) {
    int e = blockIdx.x * blockDim.x + threadIdx.x;
    if (e < E) atomicAdd(&deg[dst[e]], 1.0f);
}

__global__ void deg_finalize(float* deg, int N) {
    int i = blockIdx.x * blockDim.x + threadIdx.x;
    if (i < N) deg[i] = rsqrtf(deg[i]);
}

// ---------------- fp32 WMMA GEMM: H[N,128] = (relu?)X[N,128] @ W[128,128] ----------------
// Block = 256 threads = 8 waves. Block covers 16 rows x full 128 cols.
// Wave w computes the 16x16 tile at cols [16w,16w+16), K-loop of 32 x wmma_f32_16x16x4_f32.
__global__ void __launch_bounds__(256)
gemm128_wmma(const float* __restrict__ X, const float* __restrict__ W,
             float* __restrict__ H, int relu_in) {
    __shared__ float As[16][132];  // pad 128->132: A-frag reads hit distinct banks
    const int row0 = blockIdx.x * 16;
    const int tid  = threadIdx.x;

    // Stage 16x128 A panel into LDS (each thread: 8 floats, half-wave per row -> coalesced)
    {
        const int r = tid >> 4;          // 0..15
        const int c = (tid & 15) * 8;    // 0..120
        const float* srcp = X + (size_t)(row0 + r) * F + c;
        float4 v0 = ((const float4*)srcp)[0];
        float4 v1 = ((const float4*)srcp)[1];
        if (relu_in) {
            v0.x = fmaxf(v0.x, 0.f); v0.y = fmaxf(v0.y, 0.f);
            v0.z = fmaxf(v0.z, 0.f); v0.w = fmaxf(v0.w, 0.f);
            v1.x = fmaxf(v1.x, 0.f); v1.y = fmaxf(v1.y, 0.f);
            v1.z = fmaxf(v1.z, 0.f); v1.w = fmaxf(v1.w, 0.f);
        }
        As[r][c + 0] = v0.x; As[r][c + 1] = v0.y; As[r][c + 2] = v0.z; As[r][c + 3] = v0.w;
        As[r][c + 4] = v1.x; As[r][c + 5] = v1.y; As[r][c + 6] = v1.z; As[r][c + 7] = v1.w;
    }
    __syncthreads();

    const int lane = tid & 31;
    const int wave = tid >> 5;          // 0..7
    const int c0   = wave * 16;
    const int m    = lane & 15;         // A row / B,D col within tile
    const int kb   = (lane >> 4) * 2;   // K sub-offset for this half-wave

    v8f acc = {};
#pragma unroll
    for (int k0 = 0; k0 < F; k0 += 4) {
        // A fragment (16x4): lane = m + 16*(k>>1), vgpr = k&1
        v2f a;
        a.x = As[m][k0 + kb + 0];
        a.y = As[m][k0 + kb + 1];
        // B fragment (4x16), mirrored layout: lane = n + 16*(k>>1), vgpr = k&1
        v2f b;
        b.x = W[(size_t)(k0 + kb + 0) * F + c0 + m];
        b.y = W[(size_t)(k0 + kb + 1) * F + c0 + m];
        acc = __builtin_amdgcn_wmma_f32_16x16x4_f32(
            /*neg_a=*/false, a, /*neg_b=*/false, b,
            /*c_mod=*/(short)0, acc, /*reuse_a=*/false, /*reuse_b=*/false);
    }

    // D layout: VGPR v -> row v + 8*(lane>>4), col = lane&15
    const int rhalf = (lane >> 4) * 8;
#pragma unroll
    for (int v = 0; v < 8; ++v) {
        H[(size_t)(row0 + v + rhalf) * F + c0 + m] = acc[v];
    }
}

// ---------------- self-loop + bias: out = h * dis[i]^2 + b[f] ----------------
__global__ void selfloop_bias(const float* __restrict__ H, const float* __restrict__ dis,
                              const float* __restrict__ b, float* __restrict__ out, int NF) {
    int i = blockIdx.x * blockDim.x + threadIdx.x;
    if (i >= NF) return;
    int row = i >> 7;
    int f   = i & (F - 1);
    float s = dis[row];
    out[i] = H[i] * (s * s) + b[f];
}

// ---------------- edge scatter-add: out[dst] += h[src] * dis[src]*dis[dst] ----------------
// One wave32 per edge; lane handles float4 of the 128-wide row (512B coalesced gather).
__global__ void scatter_edges(const float* __restrict__ H, const float* __restrict__ dis,
                              const long long* __restrict__ src, const long long* __restrict__ dst,
                              float* __restrict__ out, int E) {
    int g    = blockIdx.x * blockDim.x + threadIdx.x;
    int e    = g >> 5;
    int lane = g & 31;
    if (e >= E) return;
    long long s = src[e];
    long long d = dst[e];
    float norm = dis[s] * dis[d];
    float4 v = ((const float4*)(H + (size_t)s * F))[lane];
    float* o = out + (size_t)d * F + lane * 4;
    atomicAdd(o + 0, v.x * norm);
    atomicAdd(o + 1, v.y * norm);
    atomicAdd(o + 2, v.z * norm);
    atomicAdd(o + 3, v.w * norm);
}

// ---------------- decoder: sigmoid(dot(z[src], z[dst])) ----------------
__global__ void decode_edges(const float* __restrict__ Z,
                             const long long* __restrict__ src, const long long* __restrict__ dst,
                             float* __restrict__ out, int E) {
    int g    = blockIdx.x * blockDim.x + threadIdx.x;
    int e    = g >> 5;
    int lane = g & 31;
    if (e >= E) return;
    long long s = src[e];
    long long d = dst[e];
    float4 a = ((const float4*)(Z + (size_t)s * F))[lane];
    float4 b = ((const float4*)(Z + (size_t)d * F))[lane];
    float sum = a.x * b.x + a.y * b.y + a.z * b.z + a.w * b.w;
#pragma unroll
    for (int o = 16; o > 0; o >>= 1) sum += __shfl_down(sum, o, 32);
    if (lane == 0) out[e] = 1.0f / (1.0f + __expf(-sum));
}

extern "C" void kernel_launch(void* const* d_in, const int* in_sizes, int n_in,
                              void* d_out, int out_size, void* d_ws, size_t ws_size,
                              hipStream_t stream) {
    const float*     x  = (const float*)d_in[0];
    const long long* ei = (const long long*)d_in[1];
    const float* W1 = (const float*)d_in[2];
    const float* b1 = (const float*)d_in[3];
    const float* W2 = (const float*)d_in[4];
    const float* b2 = (const float*)d_in[5];
    const float* W3 = (const float*)d_in[6];
    const float* b3 = (const float*)d_in[7];

    const int N  = in_sizes[0] / F;   // 50000
    const int E  = in_sizes[1] / 2;   // 800000
    const int NF = N * F;

    const long long* src = ei;        // edge_index[0]
    const long long* dst = ei + E;    // edge_index[1]

    // workspace layout: dis[N] | bufA[N*F] | bufB[N*F]  (~51.4 MB)
    char*  ws   = (char*)d_ws;
    float* dis  = (float*)ws;
    size_t off  = (((size_t)N * 4) + 255) & ~(size_t)255;
    float* bufA = (float*)(ws + off);
    float* bufB = bufA + (size_t)NF;

    const int T = 256;
    dim3 blk(T);
    const int gN   = (N + T - 1) / T;
    const int gE   = (E + T - 1) / T;
    const int gNF  = (NF + T - 1) / T;
    const int gE32 = (int)(((size_t)E * 32 + T - 1) / T);
    const int gGem = N / 16;          // 3125, exact -> EXEC all-ones for WMMA

    // degrees -> deg_inv_sqrt
    deg_init<<<gN, blk, 0, stream>>>(dis, N);
    deg_count<<<gE, blk, 0, stream>>>(dst, dis, E);
    deg_finalize<<<gN, blk, 0, stream>>>(dis, N);

    // layer 1: h = x@W1 ; agg = selfloop+bias + scatter ; (relu fused into next gemm)
    gemm128_wmma<<<gGem, blk, 0, stream>>>(x, W1, bufA, 0);
    selfloop_bias<<<gNF, blk, 0, stream>>>(bufA, dis, b1, bufB, NF);
    scatter_edges<<<gE32, blk, 0, stream>>>(bufA, dis, src, dst, bufB, E);

    // layer 2 (input = relu(bufB))
    gemm128_wmma<<<gGem, blk, 0, stream>>>(bufB, W2, bufA, 1);
    selfloop_bias<<<gNF, blk, 0, stream>>>(bufA, dis, b2, bufB, NF);
    scatter_edges<<<gE32, blk, 0, stream>>>(bufA, dis, src, dst, bufB, E);

    // layer 3 (input = relu(bufB)); z = bufB
    gemm128_wmma<<<gGem, blk, 0, stream>>>(bufB, W3, bufA, 1);
    selfloop_bias<<<gNF, blk, 0, stream>>>(bufA, dis, b3, bufB, NF);
    scatter_edges<<<gE32, blk, 0, stream>>>(bufA, dis, src, dst, bufB, E);

    // decode
    decode_edges<<<gE32, blk, 0, stream>>>(bufB, src, dst, (float*)d_out, E);
}